// SimpleGiMessageLayer_28003186770214
// MI455X (gfx1250) — compile-verified
//
#include <hip/hip_runtime.h>
#include <hip/hip_bf16.h>

#define N_NODES 100000
#define N_EDGES 800000
#define D_NODE  64
#define D_OUT   64

typedef __attribute__((ext_vector_type(2))) float v2f;
typedef __attribute__((ext_vector_type(8))) float v8f;

// ---------------------------------------------------------------------------
// Kernel 1: zero the per-node degree counters (d_ws is poisoned 0xAA).
// ---------------------------------------------------------------------------
__global__ void gi_zero_counts(int* __restrict__ cnt) {
    int i = blockIdx.x * blockDim.x + threadIdx.x;
    if (i < N_NODES) cnt[i] = 0;
}

// ---------------------------------------------------------------------------
// Kernel 2: histogram of node0 indices (row 0 of edge_node_indices [2, E],
// int64 per the reference). atomicAdd result unused -> no-return atomic.
// ---------------------------------------------------------------------------
__global__ void gi_count_deg(const long long* __restrict__ node0_idx,
                             int* __restrict__ cnt) {
    int e = blockIdx.x * blockDim.x + threadIdx.x;
    if (e < N_EDGES) {
        int v = (int)node0_idx[e];
        atomicAdd(&cnt[v], 1);
    }
}

// ---------------------------------------------------------------------------
// Kernel 3: z[v,:] = deg(v) * (node_features[v,:] @ W_n^T + b_n)
// One wave32 computes a 16-row x 64-col tile via V_WMMA_F32_16X16X4_F32.
//   A tile: 16x4 fp32 of node_features rows
//   B tile: 4x16 fp32 = W_n^T columns (B[k][n] = W_n[n][k])
//   4 accumulators (N-tiles of 16) x 16 K-steps of 4.
// ---------------------------------------------------------------------------
__global__ __launch_bounds__(256)
void gi_gemm_scale(const float* __restrict__ nf,
                   const float* __restrict__ Wn,
                   const float* __restrict__ bn,
                   const int*   __restrict__ cnt,
                   float*       __restrict__ z) {
    const int wave = threadIdx.x >> 5;              // 0..7
    const int lane = threadIdx.x & 31;
    const int tile = blockIdx.x * 8 + wave;         // 16-row tile index
    if (tile >= (N_NODES / 16)) return;             // wave-uniform: EXEC stays full
    const int row0 = tile * 16;
    const int mr   = lane & 15;
    const int half = lane >> 4;                     // 0 for lanes 0-15, 1 for 16-31

    v8f acc0 = {}, acc1 = {}, acc2 = {}, acc3 = {};

    // A fragment base: lanes 0-15 -> M=lane, K={k0,k0+1}; lanes 16-31 -> M=lane-16, K={k0+2,k0+3}
    const float* arow = nf + (size_t)(row0 + mr) * D_NODE + 2 * half;
    // B fragment bases: B[k][n] = Wn[n][k], n = jt*16 + mr, same K split as A
    const float* b0p = Wn + (size_t)( 0 + mr) * D_NODE + 2 * half;
    const float* b1p = Wn + (size_t)(16 + mr) * D_NODE + 2 * half;
    const float* b2p = Wn + (size_t)(32 + mr) * D_NODE + 2 * half;
    const float* b3p = Wn + (size_t)(48 + mr) * D_NODE + 2 * half;

    #pragma unroll
    for (int k = 0; k < 16; ++k) {
        const int k0 = 4 * k;
        v2f a  = *(const v2f*)(arow + k0);
        v2f b0 = *(const v2f*)(b0p + k0);
        v2f b1 = *(const v2f*)(b1p + k0);
        v2f b2 = *(const v2f*)(b2p + k0);
        v2f b3 = *(const v2f*)(b3p + k0);
        // 8 args: (neg_a, A, neg_b, B, c_mod, C, reuse_a, reuse_b)
        acc0 = __builtin_amdgcn_wmma_f32_16x16x4_f32(false, a, false, b0, (short)0, acc0, false, false);
        acc1 = __builtin_amdgcn_wmma_f32_16x16x4_f32(false, a, false, b1, (short)0, acc1, false, false);
        acc2 = __builtin_amdgcn_wmma_f32_16x16x4_f32(false, a, false, b2, (short)0, acc2, false, false);
        acc3 = __builtin_amdgcn_wmma_f32_16x16x4_f32(false, a, false, b3, (short)0, acc3, false, false);
    }

    // Epilogue: D VGPR i -> row = row0 + i + 8*half, col = jt*16 + mr
    const float bn0 = bn[ 0 + mr];
    const float bn1 = bn[16 + mr];
    const float bn2 = bn[32 + mr];
    const float bn3 = bn[48 + mr];
    #pragma unroll
    for (int i = 0; i < 8; ++i) {
        const int row = row0 + i + 8 * half;
        const float d = (float)cnt[row];
        float* zr = z + (size_t)row * D_OUT;
        zr[ 0 + mr] = (acc0[i] + bn0) * d;
        zr[16 + mr] = (acc1[i] + bn1) * d;
        zr[32 + mr] = (acc2[i] + bn2) * d;
        zr[48 + mr] = (acc3[i] + bn3) * d;
    }
}

// ---------------------------------------------------------------------------
// Inputs (setup_inputs order):
//  0 node_features f32 [100000,64]   3 W_n f32 [64,64]   4 b_n f32 [64]
//  1 edge_node_indices i64 [2,800000]  (2,5..8 unused: softmax over [E,1] == 1)
// Output: z f32 [100000,64]
// Workspace: 100000 * 4 B of degree counters.
// ---------------------------------------------------------------------------
extern "C" void kernel_launch(void* const* d_in, const int* in_sizes, int n_in,
                              void* d_out, int out_size, void* d_ws, size_t ws_size,
                              hipStream_t stream) {
    const float*     nf    = (const float*)d_in[0];
    const long long* edges = (const long long*)d_in[1];   // row 0 = node0 indices
    const float*     Wn    = (const float*)d_in[3];
    const float*     bn    = (const float*)d_in[4];
    float* z   = (float*)d_out;
    int*   cnt = (int*)d_ws;

    gi_zero_counts<<<(N_NODES + 255) / 256, 256, 0, stream>>>(cnt);
    gi_count_deg<<<(N_EDGES + 255) / 256, 256, 0, stream>>>(edges, cnt);

    const int tiles  = N_NODES / 16;          // 6250 (exact)
    const int blocks = (tiles + 7) / 8;       // 8 waves (tiles) per 256-thread block
    gi_gemm_scale<<<blocks, 256, 0, stream>>>(nf, Wn, bn, cnt, z);
}